// MemoryEnergy_29403346108722
// MI455X (gfx1250) — compile-verified
//
#include <hip/hip_runtime.h>
#include <hip/hip_bf16.h>

typedef float v2f __attribute__((ext_vector_type(2)));
typedef float v8f __attribute__((ext_vector_type(8)));

#define H_HEADS 8
#define D_HEAD  64
#define K_SLOTS 64

__global__ void me_zero_kernel(float* __restrict__ out, int n) {
    int i = blockIdx.x * blockDim.x + threadIdx.x;
    if (i < n) out[i] = 0.0f;
}

// One wave32 per (16-node tile, head).
//  - A frag (16x4 f32): lane<16 -> M=lane, d = 4k+{0,1}; lane>=16 -> M=lane-16, d = 4k+{2,3}
//  - B frag (4x16 f32): lane<16 -> N=lane, d = 4k+{0,1}; lane>=16 -> N=lane-16, d = 4k+{2,3}
//  - C (16x16 f32): VGPR r holds M=r (lanes 0-15) / M=r+8 (lanes 16-31), N = lane%16
__global__ __launch_bounds__(256)
void MemoryEnergy_29403346108722_kernel(const float* __restrict__ Qm,
                                        const float* __restrict__ Km,
                                        const int*   __restrict__ batch,
                                        const float* __restrict__ lambda_p,
                                        const float* __restrict__ beta_p,
                                        float* __restrict__ out,
                                        int n_nodes, int n_tiles) {
    const int lane = threadIdx.x & 31;
    const int wave = threadIdx.x >> 5;
    const long task    = (long)blockIdx.x * (blockDim.x >> 5) + wave;
    const long n_tasks = (long)n_tiles * H_HEADS;
    if (task >= n_tasks) return;          // wave-uniform: EXEC stays all-ones for WMMA

    const int head = (int)(task % H_HEADS);
    const int tile = (int)(task / H_HEADS);
    const int base = tile * 16;

    const int m  = lane & 15;   // row (A) / col (B) index within fragment
    const int hi = lane >> 4;   // selects K sub-pair {0,1} vs {2,3}

    const float beta   = *beta_p;
    const float coef   = (*lambda_p) / beta;
    const float zscale = beta * 0.125f;   // beta / sqrt(D_HEAD), D=64

    // ---------------- A fragments: Q tile [16 nodes x 64 d] ----------------
    const int node_a  = base + m;
    const int node_cl = node_a < n_nodes ? node_a : (n_nodes - 1);
    const float* qrow = Qm + ((long)node_cl * H_HEADS + head) * D_HEAD + 2 * hi;

    v2f a[16];
    #pragma unroll
    for (int k = 0; k < 16; ++k)
        a[k] = *(const v2f*)(qrow + 4 * k);

    // ---------------- GEMM: 16 k-steps x 4 slot tiles of v_wmma_f32_16x16x4_f32
    // Km row for this lane's slot column; slot tile t adds t*16 rows = t*1024 floats.
    const float* krow = Km + ((long)head * K_SLOTS + m) * D_HEAD + 2 * hi;

    v8f acc0 = {0.f,0.f,0.f,0.f,0.f,0.f,0.f,0.f};
    v8f acc1 = acc0, acc2 = acc0, acc3 = acc0;

    #pragma unroll
    for (int k = 0; k < 16; ++k) {
        v2f b0 = *(const v2f*)(krow +    0 + 4 * k);
        v2f b1 = *(const v2f*)(krow + 1024 + 4 * k);
        v2f b2 = *(const v2f*)(krow + 2048 + 4 * k);
        v2f b3 = *(const v2f*)(krow + 3072 + 4 * k);
        acc0 = __builtin_amdgcn_wmma_f32_16x16x4_f32(false, a[k], false, b0, (short)0, acc0, false, false);
        acc1 = __builtin_amdgcn_wmma_f32_16x16x4_f32(false, a[k], false, b1, (short)0, acc1, false, false);
        acc2 = __builtin_amdgcn_wmma_f32_16x16x4_f32(false, a[k], false, b2, (short)0, acc2, false, false);
        acc3 = __builtin_amdgcn_wmma_f32_16x16x4_f32(false, a[k], false, b3, (short)0, acc3, false, false);
    }

    // ---------------- logsumexp over 64 slots + segment scatter ----------------
    // Per lane: batch id of node base+m (clamped); sorted batch -> uniform fast path.
    const int  my_b    = batch[node_cl];
    const int  b_first = __shfl(my_b, 0);
    const int  b_last  = __shfl(my_b, 15);
    const bool full    = (base + 15) < n_nodes;
    const bool uniform = full && (b_first == b_last);

    float tile_sum = 0.0f;

    #pragma unroll
    for (int r = 0; r < 8; ++r) {
        float z0 = zscale * acc0[r];
        float z1 = zscale * acc1[r];
        float z2 = zscale * acc2[r];
        float z3 = zscale * acc3[r];

        float mx = fmaxf(fmaxf(z0, z1), fmaxf(z2, z3));
        #pragma unroll
        for (int off = 1; off <= 8; off <<= 1)
            mx = fmaxf(mx, __shfl_xor(mx, off));   // reduce over N within 16-lane group

        float s = __expf(z0 - mx) + __expf(z1 - mx) + __expf(z2 - mx) + __expf(z3 - mx);
        #pragma unroll
        for (int off = 1; off <= 8; off <<= 1)
            s += __shfl_xor(s, off);

        float lse = mx + __logf(s);   // value for node base + r + 8*hi (same in all group lanes)

        if (uniform) {
            tile_sum += lse;          // lane 0 accumulates nodes r, lane 16 nodes r+8
        } else {
            int node = base + r + 8 * hi;
            int g    = __shfl(my_b, r + 8 * hi);   // batch id of that node
            if ((lane == 0 || lane == 16) && node < n_nodes)
                atomicAdd(out + (long)g * H_HEADS + head, coef * lse);
        }
    }

    if (uniform) {
        float tot = tile_sum + __shfl_xor(tile_sum, 16);  // combine the two half-wave node sets
        if (lane == 0)
            atomicAdd(out + (long)b_first * H_HEADS + head, coef * tot);
    }
}

extern "C" void kernel_launch(void* const* d_in, const int* in_sizes, int n_in,
                              void* d_out, int out_size, void* d_ws, size_t ws_size,
                              hipStream_t stream) {
    const float* Qm       = (const float*)d_in[0];
    const float* Km       = (const float*)d_in[1];
    const int*   batch    = (const int*)d_in[2];
    // d_in[3] = num_graphs (unused on device; out_size == num_graphs * H)
    const float* lambda_p = (const float*)d_in[4];
    const float* beta_p   = (const float*)d_in[5];
    float* out = (float*)d_out;

    const int n_nodes = in_sizes[0] / (H_HEADS * D_HEAD);
    const int n_tiles = (n_nodes + 15) / 16;

    // Harness poisons d_out; we accumulate with atomics, so zero it first.
    me_zero_kernel<<<(out_size + 255) / 256, 256, 0, stream>>>(out, out_size);

    const long n_tasks = (long)n_tiles * H_HEADS;
    const int  waves_per_block = 256 / 32;
    const long n_blocks = (n_tasks + waves_per_block - 1) / waves_per_block;

    MemoryEnergy_29403346108722_kernel<<<dim3((unsigned)n_blocks), 256, 0, stream>>>(
        Qm, Km, batch, lambda_p, beta_p, out, n_nodes, n_tiles);
}